// Encoder_18468359373464
// MI455X (gfx1250) — compile-verified
//
#include <hip/hip_runtime.h>

// ---------------------------------------------------------------------------
// GCNConv (self-loops, symmetric norm) + bias + PReLU for MI455X (gfx1250).
// D = 128 fixed. fp32 throughout (matches reference dtype).
//   h = x @ W                      -> WMMA f32 16x16x4 GEMM, W staged in LDS
//   deg/dinv                       -> f32 atomics + v_rsq
//   out = sum_{e:dst=i} h[src]*n   -> wave-per-edge gather + f32 HW atomics
//   out = prelu(out + b)           -> epilogue
// ---------------------------------------------------------------------------

typedef float v2f __attribute__((ext_vector_type(2)));
typedef float v8f __attribute__((ext_vector_type(8)));

#define D 128

// ---- degree / rsqrt ---------------------------------------------------------

__global__ __launch_bounds__(256) void deg_init_kernel(float* __restrict__ deg, int N) {
    int i = blockIdx.x * 256 + threadIdx.x;
    if (i < N) deg[i] = 1.0f;  // self loop contributes 1 to every node
}

__global__ __launch_bounds__(256) void deg_accum_kernel(const long long* __restrict__ edst,
                                                        float* __restrict__ deg, int E) {
    int e = blockIdx.x * 256 + threadIdx.x;
    if (e < E) {
        int d = (int)edst[e];
        __hip_atomic_fetch_add(&deg[d], 1.0f, __ATOMIC_RELAXED, __HIP_MEMORY_SCOPE_AGENT);
    }
}

__global__ __launch_bounds__(256) void deg_to_dinv_kernel(float* __restrict__ deg, int N) {
    int i = blockIdx.x * 256 + threadIdx.x;
    if (i < N) deg[i] = __frsqrt_rn(deg[i]);  // deg >= 1 always (self loops)
}

// ---- h = x @ W via V_WMMA_F32_16X16X4_F32 ----------------------------------
// Block: 256 threads = 8 waves; block covers 128 rows of x. Each wave owns a
// 16-row stripe and all 8 column tiles (16x16), accumulating K=128 in steps
// of 4 (32 x 8 WMMAs).
//
// W is staged in LDS in K-PAIR-INTERLEAVED layout:
//   lwp[p*128 + n] = float2{ W[2p][n], W[2p+1][n] },  p = 0..63
// so a B fragment (ISA 7.12.2: lanes 0-15 need K pair {k0,k0+1}, lanes 16-31
// need {k0+2,k0+3}) is ONE ds_load_b64 into an even-aligned VGPR pair —
// no cross-register shuffles.
//
// A 16x4 f32 fragment: lane(0..15) holds A[m][k0+0..1], lane(16..31) holds
// A[m][k0+2..3]. Out-of-range rows are CLAMPED (not masked): their garbage
// only affects their own output row, which the guarded store discards. This
// keeps the inner loop branch-free (no exec-mask save/restore).

__global__ __launch_bounds__(256) void gemm_wmma_kernel(const float* __restrict__ X,
                                                        const float* __restrict__ W,
                                                        float* __restrict__ H, int N) {
    __shared__ v2f lwp[64 * D];  // 64 KB, K-pair interleaved

    const int tid = threadIdx.x;
    // cooperative fill: 8192 float2 elements, 32 per thread, coalesced in n
#pragma unroll
    for (int i = 0; i < 32; ++i) {
        int e = tid + i * 256;
        int p = e >> 7;        // K pair index
        int n = e & (D - 1);   // column
        v2f w;
        w.x = W[(2 * p) * D + n];
        w.y = W[(2 * p + 1) * D + n];
        lwp[e] = w;
    }
    __syncthreads();

    const int wave = tid >> 5;
    const int lane = tid & 31;
    const int half = lane >> 4;  // 0: K pair {k0,k0+1}, 1: K pair {k0+2,k0+3}
    const int m    = lane & 15;

    const long row0 = (long)blockIdx.x * 128 + (long)wave * 16;
    long row = row0 + m;
    if (row >= N) row = N - 1;               // clamp: branch-free inner loop
    const float* xrow = X + row * D + 2 * half;

    v8f acc[8] = {};

    for (int k0 = 0; k0 < D; k0 += 4) {
        v2f a = *(const v2f*)(xrow + k0);                   // global_load_b64
        const v2f* wbase = &lwp[(k0 >> 1) * D + half * D + m];
#pragma unroll
        for (int t = 0; t < 8; ++t) {
            v2f b = wbase[t * 16];                          // ds_load_b64
            acc[t] = __builtin_amdgcn_wmma_f32_16x16x4_f32(
                false, a, false, b, (short)0, acc[t], false, false);
        }
    }

    // store: VGPR v -> row (row0 + v + half*8), col (t*16 + m)
    float* hbase = H + (row0 + half * 8) * D + m;
    if (row0 + 16 <= (long)N) {  // full tile: unguarded stores
#pragma unroll
        for (int t = 0; t < 8; ++t)
#pragma unroll
            for (int v = 0; v < 8; ++v)
                hbase[(long)v * D + t * 16] = acc[t][v];
    } else {
#pragma unroll
        for (int t = 0; t < 8; ++t)
#pragma unroll
            for (int v = 0; v < 8; ++v)
                if (row0 + half * 8 + v < (long)N)
                    hbase[(long)v * D + t * 16] = acc[t][v];
    }
}

// ---- out initialized with self-loop term: out[i] = h[i] * dinv[i]^2 --------
// float4 per thread (4 consecutive channels of one node).

__global__ __launch_bounds__(256) void self_init_kernel(const float4* __restrict__ H,
                                                        const float* __restrict__ dinv,
                                                        float4* __restrict__ out, int total4) {
    int t = blockIdx.x * 256 + threadIdx.x;
    if (t < total4) {
        float di = dinv[t >> 5];  // node = (t*4) / 128
        float s = di * di;
        float4 hv = H[t];
        out[t] = make_float4(hv.x * s, hv.y * s, hv.z * s, hv.w * s);
    }
}

// ---- edge scatter: one wave per edge ---------------------------------------
// 32 lanes x float4 = 512B coalesced gather of h[src]; 4 hardware f32 atomic
// adds per lane into out[dst] (relaxed, agent scope -> global_atomic_add_f32).
// h (51 MB) is L2-resident (192 MB L2), so the random gather mostly hits L2.

__global__ __launch_bounds__(256) void edge_scatter_kernel(const long long* __restrict__ esrc,
                                                           const long long* __restrict__ edst,
                                                           const float* __restrict__ H,
                                                           const float* __restrict__ dinv,
                                                           float* __restrict__ out, int E) {
    int wid = (blockIdx.x * 256 + threadIdx.x) >> 5;
    if (wid >= E) return;
    int lane = threadIdx.x & 31;

    int s = (int)esrc[wid];
    int d = (int)edst[wid];
    float norm = dinv[s] * dinv[d];

    const float4 hv = *(const float4*)(H + (long)s * D + lane * 4);
    float* po = out + (long)d * D + lane * 4;
    __hip_atomic_fetch_add(po + 0, hv.x * norm, __ATOMIC_RELAXED, __HIP_MEMORY_SCOPE_AGENT);
    __hip_atomic_fetch_add(po + 1, hv.y * norm, __ATOMIC_RELAXED, __HIP_MEMORY_SCOPE_AGENT);
    __hip_atomic_fetch_add(po + 2, hv.z * norm, __ATOMIC_RELAXED, __HIP_MEMORY_SCOPE_AGENT);
    __hip_atomic_fetch_add(po + 3, hv.w * norm, __ATOMIC_RELAXED, __HIP_MEMORY_SCOPE_AGENT);
}

// ---- epilogue: out = prelu(out + b), float4 per thread ---------------------

__global__ __launch_bounds__(256) void epilogue_kernel(float4* __restrict__ out,
                                                       const float4* __restrict__ b4,
                                                       const float4* __restrict__ a4, int total4) {
    int t = blockIdx.x * 256 + threadIdx.x;
    if (t < total4) {
        int c = t & ((D / 4) - 1);  // channel group
        float4 v = out[t];
        float4 bb = b4[c];
        float4 aa = a4[c];
        v.x += bb.x; v.y += bb.y; v.z += bb.z; v.w += bb.w;
        v.x = v.x > 0.0f ? v.x : aa.x * v.x;
        v.y = v.y > 0.0f ? v.y : aa.y * v.y;
        v.z = v.z > 0.0f ? v.z : aa.z * v.z;
        v.w = v.w > 0.0f ? v.w : aa.w * v.w;
        out[t] = v;
    }
}

// ---------------------------------------------------------------------------

extern "C" void kernel_launch(void* const* d_in, const int* in_sizes, int n_in,
                              void* d_out, int out_size, void* d_ws, size_t ws_size,
                              hipStream_t stream) {
    const float*      x  = (const float*)d_in[0];
    const long long*  ei = (const long long*)d_in[1];  // int64 [2, E]
    const float*      W  = (const float*)d_in[2];
    const float*      b  = (const float*)d_in[3];
    const float*      pa = (const float*)d_in[4];

    const int N = in_sizes[0] / D;
    const int E = in_sizes[1] / 2;
    const long long* esrc = ei;
    const long long* edst = ei + E;

    float* out  = (float*)d_out;
    float* h    = (float*)d_ws;            // N*128 f32 = 51.2 MB
    float* dinv = h + (size_t)N * D;       // N f32

    const int total4 = (N * D) / 4;

    deg_init_kernel<<<(N + 255) / 256, 256, 0, stream>>>(dinv, N);
    deg_accum_kernel<<<(E + 255) / 256, 256, 0, stream>>>(edst, dinv, E);
    deg_to_dinv_kernel<<<(N + 255) / 256, 256, 0, stream>>>(dinv, N);

    gemm_wmma_kernel<<<(N + 127) / 128, 256, 0, stream>>>(x, W, h, N);

    self_init_kernel<<<(total4 + 255) / 256, 256, 0, stream>>>(
        (const float4*)h, dinv, (float4*)out, total4);
    edge_scatter_kernel<<<(E + 7) / 8, 256, 0, stream>>>(esrc, edst, h, dinv, out, E);
    epilogue_kernel<<<(total4 + 255) / 256, 256, 0, stream>>>(
        (float4*)out, (const float4*)b, (const float4*)pa, total4);
}